// Multinomial_VAE_3281355014375
// MI455X (gfx1250) — compile-verified
//
#include <hip/hip_runtime.h>
#include <hip/hip_bf16.h>
#include <float.h>

typedef __attribute__((ext_vector_type(2))) float v2f;
typedef __attribute__((ext_vector_type(8))) float v8f;

#define TAU_INV   100.0f
#define LOG_TINY  (-87.336544750f)   // logf(FLT_MIN) = log(1.17549435e-38)
#define N_TRIALS  24.0f

// ---------------------------------------------------------------------------
// GEMM: C = act(A @ B + bias).  A: MxK row-major, B: KxN row-major, fp32.
// One 256-thread block (8 waves) computes a 128x128 C tile; each wave owns a
// 32x64 sub-tile = 2x4 WMMA 16x16 accumulators (v_wmma_f32_16x16x4_f32).
// ---------------------------------------------------------------------------
template<bool RELU>
__global__ __launch_bounds__(256) void gemm_f32_wmma(
    const float* __restrict__ A, const float* __restrict__ Bm,
    const float* __restrict__ bias, float* __restrict__ C,
    int M, int Nn, int K)
{
    const int lane = threadIdx.x & 31;
    const int wave = threadIdx.x >> 5;
    const int half = lane >> 4;      // 0: lanes 0-15, 1: lanes 16-31
    const int q    = lane & 15;
    const int row0 = blockIdx.x * 128 + (wave >> 1) * 32;   // wave row base
    const int col0 = blockIdx.y * 128 + (wave & 1) * 64;    // wave col base

    const v8f vzero = {0.f,0.f,0.f,0.f,0.f,0.f,0.f,0.f};
    v8f acc[2][4];
    #pragma unroll
    for (int i = 0; i < 2; ++i)
        #pragma unroll
        for (int j = 0; j < 4; ++j) acc[i][j] = vzero;

    const float* Arow0 = A + (size_t)(row0 + q)      * K;
    const float* Arow1 = A + (size_t)(row0 + 16 + q) * K;

    #pragma unroll 4
    for (int k0 = 0; k0 < K; k0 += 4) {
        const int kr = k0 + 2 * half;       // this lane-half holds K = kr, kr+1
        // A fragments: contiguous K-pair per lane (8B aligned: kr even)
        v2f a0 = *(const v2f*)(Arow0 + kr);
        v2f a1 = *(const v2f*)(Arow1 + kr);
        // B fragments: rows kr and kr+1, coalesced across lanes
        const float* Brow = Bm + (size_t)kr * Nn + col0 + q;
        v2f b[4];
        #pragma unroll
        for (int j = 0; j < 4; ++j) {
            b[j].x = Brow[16 * j];
            b[j].y = Brow[16 * j + Nn];
        }
        #pragma unroll
        for (int j = 0; j < 4; ++j) {
            acc[0][j] = __builtin_amdgcn_wmma_f32_16x16x4_f32(
                false, a0, false, b[j], (short)0, acc[0][j], false, false);
            acc[1][j] = __builtin_amdgcn_wmma_f32_16x16x4_f32(
                false, a1, false, b[j], (short)0, acc[1][j], false, false);
        }
    }

    // Epilogue: C/D layout -> VGPR e holds M = e + 8*half, N = lane&15
    #pragma unroll
    for (int i = 0; i < 2; ++i) {
        #pragma unroll
        for (int j = 0; j < 4; ++j) {
            const int   c  = col0 + 16 * j + q;
            const float bv = bias[c];
            #pragma unroll
            for (int e = 0; e < 8; ++e) {
                const int r = row0 + 16 * i + e + 8 * half;
                float v = acc[i][j][e] + bv;
                if (RELU) v = fmaxf(v, 0.0f);
                C[(size_t)r * Nn + c] = v;
            }
        }
    }
}

// ---------------------------------------------------------------------------
// Encoder head: logits = H2 @ W3 + b3 (N=4), then log-softmax per row.
// One wave per row, shuffle butterfly reduction.
// ---------------------------------------------------------------------------
__global__ __launch_bounds__(256) void enc3_logsoftmax(
    const float* __restrict__ H2, const float* __restrict__ W3,
    const float* __restrict__ b3, float* __restrict__ logp,
    int Bsz, int Hd)
{
    const int lane = threadIdx.x & 31;
    const int wave = threadIdx.x >> 5;
    const int row  = blockIdx.x * 8 + wave;
    if (row >= Bsz) return;

    float acc[4] = {0.f, 0.f, 0.f, 0.f};
    const float* h = H2 + (size_t)row * Hd;
    for (int k = lane; k < Hd; k += 32) {
        const float hv = h[k];
        const float* w = W3 + (size_t)k * 4;
        acc[0] += hv * w[0];  acc[1] += hv * w[1];
        acc[2] += hv * w[2];  acc[3] += hv * w[3];
    }
    #pragma unroll
    for (int l = 0; l < 4; ++l)
        #pragma unroll
        for (int off = 16; off > 0; off >>= 1)
            acc[l] += __shfl_xor(acc[l], off, 32);

    float v[4], m = -FLT_MAX;
    #pragma unroll
    for (int l = 0; l < 4; ++l) { v[l] = acc[l] + b3[l]; m = fmaxf(m, v[l]); }
    float se = 0.f;
    #pragma unroll
    for (int l = 0; l < 4; ++l) se += __expf(v[l] - m);
    const float lse = __logf(se);
    if (lane < 4) logp[(size_t)row * 4 + lane] = v[lane] - m - lse;
}

// ---------------------------------------------------------------------------
// Gumbel-softmax over V count vectors -> mean_z (B x 4).
// log pmf_v = log(coeff_v) + sum_l c_vl * logp_l, clamped to [log(tiny), 0].
// s_v = (logit_v + gumbel_v)/tau; z = softmax_V(s); mean_z = (z @ cv)/N.
// Two-pass (max, then exp-sums), recomputing s_v instead of spilling it.
// ---------------------------------------------------------------------------
__global__ __launch_bounds__(256) void gumbel_mean_z(
    const float* __restrict__ logp, const float* __restrict__ rg,
    const float* __restrict__ cv, const float* __restrict__ coeffs,
    float* __restrict__ mz, int V)
{
    const int row = blockIdx.x;
    const int tid = threadIdx.x;
    __shared__ float sred[256];
    __shared__ float out5[5];

    const float lp0 = logp[row * 4 + 0], lp1 = logp[row * 4 + 1];
    const float lp2 = logp[row * 4 + 2], lp3 = logp[row * 4 + 3];
    const float* u = rg + (size_t)row * V;

    // pass 1: row max of s_v
    float lm = -FLT_MAX;
    for (int v = tid; v < V; v += 256) {
        const float* c = cv + (size_t)v * 4;
        float lg = __logf(coeffs[v]) + c[0]*lp0 + c[1]*lp1 + c[2]*lp2 + c[3]*lp3;
        lg = fminf(fmaxf(lg, LOG_TINY), 0.0f);
        const float g = -__logf(-__logf(u[v]));
        lm = fmaxf(lm, (lg + g) * TAU_INV);
    }
    sred[tid] = lm; __syncthreads();
    for (int st = 128; st > 0; st >>= 1) {
        if (tid < st) sred[tid] = fmaxf(sred[tid], sred[tid + st]);
        __syncthreads();
    }
    const float rmax = sred[0];
    __syncthreads();

    // pass 2: sum exp and weighted count-vector sums
    float se = 0.f, n0 = 0.f, n1 = 0.f, n2 = 0.f, n3 = 0.f;
    for (int v = tid; v < V; v += 256) {
        const float* c = cv + (size_t)v * 4;
        float lg = __logf(coeffs[v]) + c[0]*lp0 + c[1]*lp1 + c[2]*lp2 + c[3]*lp3;
        lg = fminf(fmaxf(lg, LOG_TINY), 0.0f);
        const float g = -__logf(-__logf(u[v]));
        const float e = __expf((lg + g) * TAU_INV - rmax);
        se += e;
        n0 += e * c[0]; n1 += e * c[1]; n2 += e * c[2]; n3 += e * c[3];
    }
    float loc[5] = {se, n0, n1, n2, n3};
    for (int qq = 0; qq < 5; ++qq) {
        sred[tid] = loc[qq]; __syncthreads();
        for (int st = 128; st > 0; st >>= 1) {
            if (tid < st) sred[tid] += sred[tid + st];
            __syncthreads();
        }
        if (tid == 0) out5[qq] = sred[0];
        __syncthreads();
    }
    if (tid == 0) {
        const float inv = 1.0f / (out5[0] * N_TRIALS);
        mz[row * 4 + 0] = out5[1] * inv;
        mz[row * 4 + 1] = out5[2] * inv;
        mz[row * 4 + 2] = out5[3] * inv;
        mz[row * 4 + 3] = out5[4] * inv;
    }
}

// ---------------------------------------------------------------------------
// Decoder first layer: H3 = relu(mean_z @ W (4xHd) + bias).  K=4 -> VALU.
// ---------------------------------------------------------------------------
__global__ __launch_bounds__(256) void dec1_relu(
    const float* __restrict__ mz, const float* __restrict__ W,
    const float* __restrict__ bias, float* __restrict__ H3, int Hd)
{
    const size_t idx = (size_t)blockIdx.x * 256 + threadIdx.x;
    const int j = (int)(idx % Hd);
    const int b = (int)(idx / Hd);
    const float m0 = mz[b*4+0], m1 = mz[b*4+1], m2 = mz[b*4+2], m3 = mz[b*4+3];
    float v = bias[j] + m0 * W[j] + m1 * W[Hd + j]
                      + m2 * W[2*Hd + j] + m3 * W[3*Hd + j];
    H3[idx] = fmaxf(v, 0.0f);
}

// ---------------------------------------------------------------------------
extern "C" void kernel_launch(void* const* d_in, const int* in_sizes, int n_in,
                              void* d_out, int out_size, void* d_ws, size_t ws_size,
                              hipStream_t stream)
{
    const float* x      = (const float*)d_in[0];
    const float* rgrid  = (const float*)d_in[1];
    const float* enc_w1 = (const float*)d_in[2];
    const float* enc_b1 = (const float*)d_in[3];
    const float* enc_w2 = (const float*)d_in[4];
    const float* enc_b2 = (const float*)d_in[5];
    const float* enc_w3 = (const float*)d_in[6];
    const float* enc_b3 = (const float*)d_in[7];
    const float* dec_w1 = (const float*)d_in[8];
    const float* dec_b1 = (const float*)d_in[9];
    const float* dec_w2 = (const float*)d_in[10];
    const float* dec_b2 = (const float*)d_in[11];
    const float* dec_w3 = (const float*)d_in[12];
    const float* dec_b3 = (const float*)d_in[13];
    const float* cv     = (const float*)d_in[14];
    const float* coeffs = (const float*)d_in[15];

    const int Bsz = 4096, Dd = 1024, Hd = 1024;
    const int V = in_sizes[15];              // 2925

    float* h1   = (float*)d_ws;              // 16 MB (reused for h3)
    float* h2   = h1 + (size_t)Bsz * Hd;     // 16 MB (reused for h4)
    float* logp = h2 + (size_t)Bsz * Hd;     // B x 4
    float* mz   = logp + (size_t)Bsz * 4;    // B x 4

    const dim3 blk(256);
    // encoder
    gemm_f32_wmma<true ><<<dim3(Bsz/128, Hd/128), blk, 0, stream>>>(x,  enc_w1, enc_b1, h1, Bsz, Hd, Dd);
    gemm_f32_wmma<true ><<<dim3(Bsz/128, Hd/128), blk, 0, stream>>>(h1, enc_w2, enc_b2, h2, Bsz, Hd, Hd);
    enc3_logsoftmax<<<Bsz/8, blk, 0, stream>>>(h2, enc_w3, enc_b3, logp, Bsz, Hd);
    // multinomial pmf + Gumbel-softmax + mean_z
    gumbel_mean_z<<<Bsz, blk, 0, stream>>>(logp, rgrid, cv, coeffs, mz, V);
    // decoder
    dec1_relu<<<(Bsz * Hd) / 256, blk, 0, stream>>>(mz, dec_w1, dec_b1, h1, Hd);
    gemm_f32_wmma<true ><<<dim3(Bsz/128, Hd/128), blk, 0, stream>>>(h1, dec_w2, dec_b2, h2, Bsz, Hd, Hd);
    gemm_f32_wmma<false><<<dim3(Bsz/128, Dd/128), blk, 0, stream>>>(h2, dec_w3, dec_b3, (float*)d_out, Bsz, Dd, Hd);
}